// tb_floquet_tbc_cuda_9783935500627
// MI455X (gfx1250) — compile-verified
//
#include <hip/hip_runtime.h>

// Problem constants (static in the reference): NX=64, NY=32, N=2048, J=1.0
#define NXC 64
#define NYC 32
#define NC  2048   // NX*NY
// Output: [4, N, N] complex64 -> 4*2048*2048*2 floats = 128 MiB

typedef float f2v __attribute__((ext_vector_type(2)));
typedef float f4v __attribute__((ext_vector_type(4)));
typedef int   v4i __attribute__((ext_vector_type(4)));

#if defined(__HIP_DEVICE_COMPILE__)
typedef __attribute__((address_space(1))) v4i g_v4i;   // global int4
typedef __attribute__((address_space(3))) v4i s_v4i;   // LDS int4
#endif

// ---------------------------------------------------------------------------
// Kernel 1: bandwidth-bound zero fill of the whole 128 MiB output.
// Preferred path: CDNA5 async LDS->global B128 stores (ASYNCcnt pipeline),
// 512 B per wave per instruction, zeros never pass through VGPR writeback.
// ---------------------------------------------------------------------------
__global__ __launch_bounds__(256) void fz_fill_kernel(f4v* __restrict__ out, long n4) {
    __shared__ f4v zlds[256];
    f4v z;
    z.x = 0.0f; z.y = 0.0f; z.z = 0.0f; z.w = 0.0f;
    zlds[threadIdx.x] = z;
    __syncthreads();

    long stride = (long)gridDim.x * (long)blockDim.x;
    long i = (long)blockIdx.x * (long)blockDim.x + (long)threadIdx.x;

#if defined(__HIP_DEVICE_COMPILE__) && __has_builtin(__builtin_amdgcn_global_store_async_from_lds_b128)
    for (; i < n4; i += stride) {
        __builtin_amdgcn_global_store_async_from_lds_b128(
            (g_v4i*)(out + i),
            (s_v4i*)&zlds[threadIdx.x],
            /*offset=*/0, /*cpol=*/0);
    }
#if __has_builtin(__builtin_amdgcn_s_wait_asynccnt)
    __builtin_amdgcn_s_wait_asynccnt(0);
#else
    asm volatile("s_wait_asynccnt 0" ::: "memory");
#endif
#else
    for (; i < n4; i += stride) {
        out[i] = z;
    }
#endif
}

// ---------------------------------------------------------------------------
// Kernel 2: tiny static scatter. One thread per Hamiltonian replays the
// reference index-pair generation exactly (NX=64 even, NY=32 even branches).
// Interior bonds (-J, 0) first, then boundary bonds (-J e^{+-i theta}),
// preserving the reference's .set override order. OOB writes dropped (JAX
// scatter drop semantics).
// ---------------------------------------------------------------------------
__device__ __forceinline__ void h_write(f2v* out, int h, int r, int c,
                                        float re, float im) {
    if ((unsigned)r < (unsigned)NC && (unsigned)c < (unsigned)NC) {
        f2v v; v.x = re; v.y = im;
        out[((size_t)h * NC + (size_t)r) * NC + (size_t)c] = v;
    }
}

__device__ __forceinline__ void h_bond(f2v* out, int h, int a, int b) {
    h_write(out, h, a, b, -1.0f, 0.0f);
    h_write(out, h, b, a, -1.0f, 0.0f);
}

__global__ void tb_scatter_kernel(f2v* __restrict__ out,
                                  const float* __restrict__ theta_x,
                                  const float* __restrict__ theta_y) {
    const int h = blockIdx.x;   // 0..3
    if (threadIdx.x != 0) return;

    const int nx = NXC, ny = NYC, n = NC;
    const float th = (h == 0 || h == 2) ? *theta_y : *theta_x;
    const float c = __cosf(th), s = __sinf(th);
    const float fre = -c, fim = -s;   // -J * exp(+i th)
    const float bre = -c, bim =  s;   // -J * exp(-i th)

    if (h == 0) {
        // ---- H1 interior (nx even) ----
        int a0 = 0, b0 = nx;
        for (int j = 0; j < nx; ++j) {
            if (j) { int inc = (j == nx / 2) ? 3 : 2; a0 += inc; b0 += inc; }
            for (int a = a0, b = b0; a < n && b < n; a += 2 * nx, b += 2 * nx)
                h_bond(out, 0, a, b);
        }
        // ---- H1 boundary (ny even), uses theta_y ----
        for (int p = 0; 1 + 2 * p < nx; ++p) {
            int ba = 1 + nx * (ny - 1) + 2 * p, bb = 1 + 2 * p;
            h_write(out, 0, ba, bb, fre, fim);
            h_write(out, 0, bb, ba, bre, bim);
        }
    } else if (h == 1) {
        // ---- H2 interior (nx even, nx != 2) ----
        int n1 = 1, a = -1, b = 0;
        while (n1 <= ny && a < n - 1 && b < n) {
            a += 2; b += 2;
            if (b < n1 * nx) {
                h_bond(out, 1, a, b);
            } else {
                n1 += 1;
                a += 1; b += 1;                 // nx even
                if (a < n - 1 && b < n) h_bond(out, 1, a, b);
            }
            if (b >= n - 2) break;
        }
        // ---- H2 boundary (nx even), uses theta_x ----
        for (int p = 0; nx - 1 + 2 * nx * p < n; ++p) {
            int ba = nx - 1 + 2 * nx * p, bb = 2 * nx * p;
            h_write(out, 1, ba, bb, fre, fim);
            h_write(out, 1, bb, ba, bre, bim);
        }
    } else if (h == 2) {
        // ---- H3 interior (nx even) ----
        int n1 = 1, a = 1, b = 1 + nx;
        if (b < n) h_bond(out, 2, a, b);
        while (n1 < ny && a < n - 1 && b < n - 1) {
            a += 2; b += 2;
            if (a < n1 * nx) {
                h_bond(out, 2, a, b);
            } else {
                n1 += 1;
                if (n1 % 2 == 0)      { a -= 1; b -= 1; }
                else if (b < n - 1)   { a += 1; b += 1; }
                else                  { a -= 2; b -= 2; }
                h_bond(out, 2, a, b);
            }
        }
        // ---- H3 boundary (ny even), uses theta_y ----
        for (int p = 0; 2 * p < nx; ++p) {
            int ba = nx * (ny - 1) + 2 * p, bb = 2 * p;
            h_write(out, 2, ba, bb, fre, fim);
            h_write(out, 2, bb, ba, bre, bim);
        }
    } else {
        // ---- H4 interior (nx even, nx != 2) ----
        int n1 = 1, a = -2, b = -1;
        while (n1 <= ny && a < n - 2 && b < n - 2) {
            a += 2; b += 2;
            if (b < n1 * nx) {
                h_bond(out, 3, a, b);
            } else {
                n1 += 1;
                a += 1; b += 1;                 // nx even, nx != 2
                h_bond(out, 3, a, b);
            }
        }
        // ---- H4 boundary (nx even), uses theta_x ----
        for (int p = 0; 2 * nx * (1 + p) - 1 < n; ++p) {
            int ba = 2 * nx * (1 + p) - 1, bb = 2 * nx * p;
            h_write(out, 3, ba, bb, fre, fim);
            h_write(out, 3, bb, ba, bre, bim);
        }
    }
}

extern "C" void kernel_launch(void* const* d_in, const int* in_sizes, int n_in,
                              void* d_out, int out_size, void* d_ws, size_t ws_size,
                              hipStream_t stream) {
    (void)in_sizes; (void)n_in; (void)d_ws; (void)ws_size;
    const float* theta_x = (const float*)d_in[0];
    const float* theta_y = (const float*)d_in[1];

    // out_size = 4*2048*2048*2 floats; fill as float4 chunks.
    long n4 = (long)out_size / 4;
    fz_fill_kernel<<<2048, 256, 0, stream>>>((f4v*)d_out, n4);
    tb_scatter_kernel<<<4, 1, 0, stream>>>((f2v*)d_out, theta_x, theta_y);
}